// Sparsemax_1675037245863
// MI455X (gfx1250) — compile-verified
//
#include <hip/hip_runtime.h>
#include <stdint.h>

// Row-wise sparsemax for X[4096][32000] fp32.
// Memory-bound: 1 GB traffic -> ~44 us floor @ 23.3 TB/s.
// Per block: TDM-DMA one 128KB row into LDS (NT hint: pure streaming, no reuse,
// don't thrash the 192MB L2), pull to VGPRs, candidate-filter (support subset
// of {x > rowmax-1}), exact Michelot fixpoint on the tiny candidate set,
// streaming nontemporal b128 stores of max((x-max)-tau, 0).

#define ROW_D   32000
#define ROW_D4  8000          // row length in float4
#define NTHR    1024          // 32 wave32 waves per block
#define NWAVE   32
#define EPT4    8             // float4 per thread (8*1024 >= 8000)

typedef unsigned int u32x4 __attribute__((ext_vector_type(4)));
typedef int          i32x4 __attribute__((ext_vector_type(4)));
typedef int          i32x8 __attribute__((ext_vector_type(8)));
typedef float        f32x4 __attribute__((ext_vector_type(4)));

struct Pair { float s; float c; };

// Block-wide (sum,sum) reduction. Leading barrier protects scratch reuse,
// trailing barrier publishes partials. Fixed summation order -> deterministic.
__device__ __forceinline__ Pair blockSum2(float a, float b, float* sA, float* sB) {
  const int t = threadIdx.x;
  #pragma unroll
  for (int o = 16; o > 0; o >>= 1) {
    a += __shfl_xor(a, o, 32);
    b += __shfl_xor(b, o, 32);
  }
  __syncthreads();
  if ((t & 31) == 0) { sA[t >> 5] = a; sB[t >> 5] = b; }
  __syncthreads();
  float ra = 0.0f, rb = 0.0f;
  #pragma unroll
  for (int w = 0; w < NWAVE; ++w) { ra += sA[w]; rb += sB[w]; }
  Pair p; p.s = ra; p.c = rb; return p;
}

__global__ __launch_bounds__(NTHR) void sparsemax_rows(const float* __restrict__ X,
                                                       float* __restrict__ Y,
                                                       int nrows) {
  __shared__ float4 srow[ROW_D4];   // 128 KB row staging (TDM destination)
  __shared__ float  sA[NWAVE];
  __shared__ float  sB[NWAVE];
  __shared__ float  scand[NTHR];    // shifted candidate values

  const int t    = threadIdx.x;
  const int lane = t & 31;
  const int wid  = t >> 5;
  const int row  = blockIdx.x;
  if (row >= nrows) return;

  float4 v[EPT4];

#if __has_builtin(__builtin_amdgcn_tensor_load_to_lds)
  // ---- Tensor Data Mover: DMA the whole 128 KB row global -> LDS ----
  if (t < 32) {   // wave 0 only issues the wave-level DMA op
    uint64_t ga   = (uint64_t)(uintptr_t)(X + (size_t)row * ROW_D);
    uint32_t lds  = (uint32_t)(uintptr_t)(&srow[0]);   // generic low32 == LDS byte offset
    uint32_t galo = (uint32_t)__builtin_amdgcn_readfirstlane((int)(uint32_t)ga);
    uint32_t gahi = (uint32_t)__builtin_amdgcn_readfirstlane((int)(uint32_t)(ga >> 32));
    uint32_t ldsu = (uint32_t)__builtin_amdgcn_readfirstlane((int)lds);

    // D# group 0: count=1 | lds_addr | global_addr[56:0] | type=2
    u32x4 g0 = { 1u, ldsu, galo, (gahi & 0x01FFFFFFu) | 0x80000000u };
    // D# group 1: data_size=4B; tensor_dim0=32000; tensor_dim1=1;
    //             tile_dim0=32000; tile_dim1=1; tensor_dim0_stride=32000
    i32x8 g1 = { 0x00020000,
                 (int)((ROW_D & 0xFFFF) << 16),
                 (int)(((ROW_D >> 16) & 0xFFFF) | (1 << 16)),
                 (int)((ROW_D & 0xFFFF) << 16),
                 1,
                 ROW_D,
                 0, 0 };
    i32x4 z4 = {0, 0, 0, 0};
#if __clang_major__ >= 23
    i32x8 z8 = {0, 0, 0, 0, 0, 0, 0, 0};
    __builtin_amdgcn_tensor_load_to_lds(g0, g1, z4, z4, z8, 1);  // cpol: TH=NT (streaming)
#else
    __builtin_amdgcn_tensor_load_to_lds(g0, g1, z4, z4, 1);      // cpol: TH=NT (streaming)
#endif
#if __has_builtin(__builtin_amdgcn_s_wait_tensorcnt)
    __builtin_amdgcn_s_wait_tensorcnt(0);
#else
    asm volatile("s_wait_tensorcnt 0x0" ::: "memory");
#endif
  }
  __syncthreads();  // publish DMA'd row to all 32 waves

  #pragma unroll
  for (int i = 0; i < EPT4; ++i) {
    const int idx4 = i * NTHR + t;
    v[i] = (idx4 < ROW_D4)
             ? srow[idx4]
             : make_float4(-__builtin_inff(), -__builtin_inff(),
                           -__builtin_inff(), -__builtin_inff());
  }
#else
  // Toolchain without TDM builtin: direct vectorized global loads.
  const float4* rowp = (const float4*)(X + (size_t)row * ROW_D);
  #pragma unroll
  for (int i = 0; i < EPT4; ++i) {
    const int idx4 = i * NTHR + t;
    if (idx4 < ROW_D4) v[i] = rowp[idx4];
    else v[i] = make_float4(-1e30f, -1e30f, -1e30f, -1e30f);
  }
#endif

  // ---- Pass 1: row max ----
  float mx = -__builtin_inff();
  #pragma unroll
  for (int i = 0; i < EPT4; ++i)
    mx = fmaxf(mx, fmaxf(fmaxf(v[i].x, v[i].y), fmaxf(v[i].z, v[i].w)));
  #pragma unroll
  for (int o = 16; o > 0; o >>= 1) mx = fmaxf(mx, __shfl_xor(mx, o, 32));
  if (lane == 0) sA[wid] = mx;
  __syncthreads();
  float rowmax = sA[0];
  #pragma unroll
  for (int w = 1; w < NWAVE; ++w) rowmax = fmaxf(rowmax, sA[w]);
  __syncthreads();

  // ---- Pass 2: candidate filter. tau >= rowmax-1, so support ⊆ {x > rowmax-1}.
  // Slack makes the set a superset under fp rounding (harmless for Michelot).
  const float thr0 = (rowmax - 1.0f) - fabsf(rowmax) * 2e-6f - 2e-6f;
  int cnt = 0;
  #pragma unroll
  for (int i = 0; i < EPT4; ++i)
    cnt += (v[i].x > thr0) + (v[i].y > thr0) + (v[i].z > thr0) + (v[i].w > thr0);

  // Deterministic compaction: wave inclusive scan + wave-total scan.
  int inc = cnt;
  #pragma unroll
  for (int o = 1; o < 32; o <<= 1) {
    int y = __shfl_up(inc, o, 32);
    if (lane >= o) inc += y;
  }
  if (lane == 31) sB[wid] = (float)inc;
  __syncthreads();
  int waveBase = 0, total = 0;
  #pragma unroll
  for (int w = 0; w < NWAVE; ++w) {
    const int c = (int)sB[w];
    total += c;
    if (w < wid) waveBase += c;
  }
  int base = waveBase + (inc - cnt);   // exclusive prefix for this thread
  __syncthreads();

  if (total <= NTHR) {
    #pragma unroll
    for (int i = 0; i < EPT4; ++i) {
      const float comp[4] = { v[i].x, v[i].y, v[i].z, v[i].w };
      #pragma unroll
      for (int j = 0; j < 4; ++j)
        if (comp[j] > thr0) scand[base++] = comp[j] - rowmax;  // shifted, in [-1,0]
    }
  }
  __syncthreads();

  // ---- Solve for tau (shifted space) ----
  float tau;
  if (total <= NTHR) {
    // Michelot fixpoint on the (tiny) candidate set: exact sparsemax tau.
    const bool  have = (t < total);
    const float c    = have ? scand[t] : 0.0f;
    tau = -2.0f;                              // below every candidate
    #pragma unroll 1
    for (int it = 0; it < 64; ++it) {
      const bool act = have && (c > tau);
      Pair s = blockSum2(act ? c : 0.0f, act ? 1.0f : 0.0f, sA, sB);
      const float tn = (s.s - 1.0f) / s.c;    // s.c >= 1 always (max elem, tau<0)
      if (tn == tau) break;                   // uniform across block
      tau = tn;
    }
  } else {
    // Fallback (degenerate rows): bisection on register data + exact refine.
    float lo = -1.0f, hi = 0.0f;
    #pragma unroll 1
    for (int it = 0; it < 30; ++it) {
      const float mid = 0.5f * (lo + hi);
      float ps = 0.0f;
      #pragma unroll
      for (int i = 0; i < EPT4; ++i) {
        ps += fmaxf((v[i].x - rowmax) - mid, 0.0f);
        ps += fmaxf((v[i].y - rowmax) - mid, 0.0f);
        ps += fmaxf((v[i].z - rowmax) - mid, 0.0f);
        ps += fmaxf((v[i].w - rowmax) - mid, 0.0f);
      }
      Pair s = blockSum2(ps, 0.0f, sA, sB);
      if (s.s >= 1.0f) lo = mid; else hi = mid;
    }
    float ps = 0.0f, pc = 0.0f;
    #pragma unroll
    for (int i = 0; i < EPT4; ++i) {
      const float comp[4] = { v[i].x, v[i].y, v[i].z, v[i].w };
      #pragma unroll
      for (int j = 0; j < 4; ++j) {
        const float xs = comp[j] - rowmax;
        if (xs > lo) { ps += xs; pc += 1.0f; }
      }
    }
    Pair s = blockSum2(ps, pc, sA, sB);
    tau = (s.s - 1.0f) / fmaxf(s.c, 1.0f);
  }

  // ---- Pass 3: out = max((x - rowmax) - tau, 0), nontemporal b128 stores ----
  f32x4* Yo = (f32x4*)Y + (size_t)row * ROW_D4;
  #pragma unroll
  for (int i = 0; i < EPT4; ++i) {
    const int idx4 = i * NTHR + t;
    if (idx4 < ROW_D4) {
      f32x4 o;
      o.x = fmaxf((v[i].x - rowmax) - tau, 0.0f);
      o.y = fmaxf((v[i].y - rowmax) - tau, 0.0f);
      o.z = fmaxf((v[i].z - rowmax) - tau, 0.0f);
      o.w = fmaxf((v[i].w - rowmax) - tau, 0.0f);
#if __has_builtin(__builtin_nontemporal_store)
      __builtin_nontemporal_store(o, &Yo[idx4]);
#else
      Yo[idx4] = o;
#endif
    }
  }
}

extern "C" void kernel_launch(void* const* d_in, const int* in_sizes, int n_in,
                              void* d_out, int out_size, void* d_ws, size_t ws_size,
                              hipStream_t stream) {
  (void)n_in; (void)out_size; (void)d_ws; (void)ws_size;
  const float* X = (const float*)d_in[0];
  float*       Y = (float*)d_out;
  const int nrows = in_sizes[0] / ROW_D;   // 4096
  sparsemax_rows<<<nrows, NTHR, 0, stream>>>(X, Y, nrows);
}